// GRID_Predictor_79663053406480
// MI455X (gfx1250) — compile-verified
//
#include <hip/hip_runtime.h>
#include <hip/hip_bf16.h>

// ---------------------------------------------------------------------------
// Emission field evaluation (velocity-warped trilinear grid lookup + masks).
// Memory-bound elementwise kernel: 384 MB streamed @ 23.3 TB/s => ~16.5 us
// floor on MI455X. No matmul in this op, so no WMMA; the CDNA5-specific wins
// are b128 NT-hinted streaming, global_prefetch_b8, EXEC-predicated gather
// (one mask per point, branchless cndmask taps) and the 1 MB voxel grid held
// hot in WGP$/L2 because the big streams carry non-temporal hints.
// ---------------------------------------------------------------------------

namespace {
constexpr float kScale = 10.0f;   // domain half-width
constexpr float kRMin  = 2.0f;
constexpr float kRMax  = 10.0f;
constexpr float kZW    = 4.0f;
constexpr int   kGR    = 64;      // grid resolution
constexpr int   kNT = 32, kNA = 64, kNB = 64, kNG = 128;
constexpr long long kN = (long long)kNT * kNA * kNB * kNG;   // 16,777,216
constexpr int   kV4     = (int)(kN / 4);                     // float4 packs
constexpr int   kShiftNtPack = 17;  // log2(NA*NB*NG) - log2(4) = 19 - 2
constexpr int   kBlock  = 256;      // 8 wave32 waves per workgroup
constexpr int   kGrid   = 2048;     // persistent blocks; 8 sweeps over data
constexpr float kIdxScale = (kGR - 1) / (2.0f * kScale);     // (v+S)*this
}

typedef float f4 __attribute__((ext_vector_type(4)));

__device__ __forceinline__ int clamp63(int v) {
  v = v < 0 ? 0 : v;
  return v > (kGR - 1) ? (kGR - 1) : v;
}

__global__ void __launch_bounds__(kBlock)
emission_kernel(const float* __restrict__ t_frames,
                const float* __restrict__ coords,
                const float* __restrict__ Omega,
                const float* __restrict__ t_geos,
                const float* __restrict__ t_injection,
                const float* __restrict__ t_start_obs,
                const float* __restrict__ grid,
                float* __restrict__ out) {
  const float tinj   = t_injection[0];
  const float tstart = t_start_obs[0];

  const f4* __restrict__ xp = (const f4*)(coords);
  const f4* __restrict__ yp = (const f4*)(coords + kN);
  const f4* __restrict__ zp = (const f4*)(coords + 2 * kN);
  const f4* __restrict__ op = (const f4*)(Omega);
  const f4* __restrict__ gp = (const f4*)(t_geos);
  f4* __restrict__ outp     = (f4*)(out);

  const int stride = gridDim.x * blockDim.x;  // 524288 threads / sweep
  for (int i = blockIdx.x * blockDim.x + threadIdx.x; i < kV4; i += stride) {
    // Prefetch next sweep's cachelines (speculative, global_prefetch_b8).
    const int pf = i + stride;
    if (pf < kV4) {
      __builtin_prefetch((const void*)(xp + pf), 0, 1);
      __builtin_prefetch((const void*)(yp + pf), 0, 1);
      __builtin_prefetch((const void*)(zp + pf), 0, 1);
      __builtin_prefetch((const void*)(op + pf), 0, 1);
      __builtin_prefetch((const void*)(gp + pf), 0, 1);
    }

    // Non-temporal b128 streaming loads: don't pollute WGP$/L2 (keep the
    // 1 MB voxel grid resident for the gather phase).
    f4 x  = __builtin_nontemporal_load(xp + i);
    f4 y  = __builtin_nontemporal_load(yp + i);
    f4 z  = __builtin_nontemporal_load(zp + i);
    f4 om = __builtin_nontemporal_load(op + i);
    f4 tg = __builtin_nontemporal_load(gp + i);

    // NA*NB*NG = 2^19 is a multiple of 4, so one frame index per pack.
    const int   nt = i >> kShiftNtPack;
    const float tM = t_frames[nt] - tstart;

    f4 res;
#pragma unroll
    for (int j = 0; j < 4; ++j) {
      const float xx = x[j], yy = y[j], zz = z[j];
      const float t_rot = tM - tg[j] - tinj;
      const float r2 = xx * xx + yy * yy + zz * zz;

      // All zeroing masks from the reference folded into one predicate:
      //   t_rot < 0 (invalid/inf sentinel), r outside [RMIN,RMAX], |z| > ZW.
      const bool pass = (t_rot >= 0.0f) &&
                        (r2 >= kRMin * kRMin) && (r2 <= kRMax * kRMax) &&
                        (__builtin_fabsf(zz) <= kZW);

      float e = 0.0f;
      if (pass) {  // single EXEC mask; masked lanes issue no gather traffic
        float s, c;
        __sincosf(-om[j] * t_rot, &s, &c);   // v_sin/v_cos trans ops
        const float xw = xx * c - yy * s;
        const float yw = xx * s + yy * c;

        const float gx = (xw + kScale) * kIdxScale;
        const float gy = (yw + kScale) * kIdxScale;
        const float gz = (zz + kScale) * kIdxScale;

        const float fx = floorf(gx), fy = floorf(gy), fz = floorf(gz);
        const int ix = (int)fx, iy = (int)fy, iz = (int)fz;
        const float wx = gx - fx, wy = gy - fy, wz = gz - fz;

        // Branchless cval=0 taps. Given pass: |z|<=4 => gz in [18.9,44.1],
        // so z taps are always in range; rotation preserves x^2+y^2 and
        // r<=10 => |xw|,|yw|<=10, so x/y masks only fire in FP edge cases.
        // Clamp addresses (memory-safe), zero via cndmask (no exec churn).
        const bool bx0 = (unsigned)ix       < (unsigned)kGR;
        const bool bx1 = (unsigned)(ix + 1) < (unsigned)kGR;
        const bool by0 = (unsigned)iy       < (unsigned)kGR;
        const bool by1 = (unsigned)(iy + 1) < (unsigned)kGR;

        const int X0 = clamp63(ix)     * (kGR * kGR);
        const int X1 = clamp63(ix + 1) * (kGR * kGR);
        const int Y0 = clamp63(iy)     * kGR;
        const int Y1 = clamp63(iy + 1) * kGR;
        const int Z0 = iz;        // provably in [18,44] under pass
        const int Z1 = iz + 1;    // provably <= 45

        float v000 = grid[X0 + Y0 + Z0];
        float v001 = grid[X0 + Y0 + Z1];
        float v010 = grid[X0 + Y1 + Z0];
        float v011 = grid[X0 + Y1 + Z1];
        float v100 = grid[X1 + Y0 + Z0];
        float v101 = grid[X1 + Y0 + Z1];
        float v110 = grid[X1 + Y1 + Z0];
        float v111 = grid[X1 + Y1 + Z1];

        v000 = (bx0 && by0) ? v000 : 0.0f;
        v001 = (bx0 && by0) ? v001 : 0.0f;
        v010 = (bx0 && by1) ? v010 : 0.0f;
        v011 = (bx0 && by1) ? v011 : 0.0f;
        v100 = (bx1 && by0) ? v100 : 0.0f;
        v101 = (bx1 && by0) ? v101 : 0.0f;
        v110 = (bx1 && by1) ? v110 : 0.0f;
        v111 = (bx1 && by1) ? v111 : 0.0f;

        // Trilinear blend (fma form), innermost (z) first.
        const float c00 = fmaf(v001 - v000, wz, v000);
        const float c01 = fmaf(v011 - v010, wz, v010);
        const float c10 = fmaf(v101 - v100, wz, v100);
        const float c11 = fmaf(v111 - v110, wz, v110);
        const float c0  = fmaf(c01 - c00, wy, c00);
        const float c1  = fmaf(c11 - c10, wy, c10);
        const float net = fmaf(c1 - c0, wx, c0);

        // sigmoid(net - 10)
        e = 1.0f / (1.0f + __expf(10.0f - net));
      }
      res[j] = e;
    }

    // Non-temporal b128 store of the emission pack.
    __builtin_nontemporal_store(res, outp + i);
  }
}

extern "C" void kernel_launch(void* const* d_in, const int* in_sizes, int n_in,
                              void* d_out, int out_size, void* d_ws,
                              size_t ws_size, hipStream_t stream) {
  // setup_inputs() order:
  //   0: t_frames (32), 1: coords (3*N), 2: Omega (N), 3: t_geos (N),
  //   4: t_injection (1), 5: t_start_obs (1), 6: grid (64^3)
  const float* t_frames    = (const float*)d_in[0];
  const float* coords      = (const float*)d_in[1];
  const float* Omega       = (const float*)d_in[2];
  const float* t_geos      = (const float*)d_in[3];
  const float* t_injection = (const float*)d_in[4];
  const float* t_start_obs = (const float*)d_in[5];
  const float* grid        = (const float*)d_in[6];
  float* out = (float*)d_out;

  emission_kernel<<<dim3(kGrid), dim3(kBlock), 0, stream>>>(
      t_frames, coords, Omega, t_geos, t_injection, t_start_obs, grid, out);
}